// ROLAND_33285996544265
// MI455X (gfx1250) — compile-verified
//
#include <hip/hip_runtime.h>
#include <hip/hip_bf16.h>
#include <math.h>

typedef float v2f __attribute__((ext_vector_type(2)));
typedef float v8f __attribute__((ext_vector_type(8)));

// ---------------------------------------------------------------------------
// FP32 WMMA GEMM: C[M,Nc] = op(A[M,K] x B) (+bias) (+relu) (+=C)
// non-trans: B[k*ldb+n] ; trans: B[n*ldb+k] (weight stored [Nc,K] row-major)
// One wave -> 5 consecutive 16-row tiles x one 16-col tile.
// REQUIRES: M % 80 == 0, Nc % 16 == 0, K % 4 == 0, grid sized exactly.
// No divergent control flow anywhere -> EXEC is all-ones for every WMMA.
// blockDim = 128 (4 waves).
// ---------------------------------------------------------------------------
template <bool TRANSB, bool BIAS, bool RELU, bool ACCUM>
__global__ __launch_bounds__(128) void gemm_f32_wmma(
    const float* __restrict__ A, int lda,
    const float* __restrict__ B, int ldb,
    const float* __restrict__ bias,
    float* __restrict__ C, int ldc,
    int K, int Ntiles)
{
    const int lane  = threadIdx.x & 31;
    const int wave  = blockIdx.x * 4 + (threadIdx.x >> 5);
    const int tileN = wave % Ntiles;
    const int mg    = wave / Ntiles;
    const int baseM = mg * 80;              // 5 tiles of 16 rows
    const int half  = lane >> 4;            // K-half select
    const int mr    = lane & 15;
    const int col   = tileN * 16 + mr;

    // Per-lane base pointers (k advances by +4 floats each step)
    const float* aptr = A + (size_t)(baseM + mr) * lda + 2 * half;
    const float* bptr;
    if (TRANSB) bptr = B + (size_t)col * ldb + 2 * half;
    else        bptr = B + (size_t)(2 * half) * ldb + col;

    v8f acc[5] = {};

#pragma unroll 2
    for (int k0 = 0; k0 < K; k0 += 4) {
        v2f bfrag;
        if (TRANSB) {
            bfrag = *reinterpret_cast<const v2f*>(bptr + k0);
        } else {
            bfrag[0] = bptr[(size_t)k0 * ldb];
            bfrag[1] = bptr[(size_t)(k0 + 1) * ldb];
        }
        v2f afrag[5];
#pragma unroll
        for (int t = 0; t < 5; ++t)
            afrag[t] = *reinterpret_cast<const v2f*>(aptr + (size_t)t * 16 * lda + k0);
#pragma unroll
        for (int t = 0; t < 5; ++t)
            acc[t] = __builtin_amdgcn_wmma_f32_16x16x4_f32(
                false, afrag[t], false, bfrag, (short)0, acc[t], false, false);
    }

    float bb = 0.0f;
    if (BIAS) bb = bias[col];
#pragma unroll
    for (int t = 0; t < 5; ++t) {
        const int rb = baseM + t * 16 + 8 * half;   // D: c[v] = row v+8*half
#pragma unroll
        for (int v = 0; v < 8; ++v) {
            float val = acc[t][v] + bb;
            if (RELU) val = fmaxf(val, 0.0f);
            const size_t idx = (size_t)(rb + v) * ldc + col;
            if (ACCUM) val += C[idx];
            C[idx] = val;
        }
    }
}

// ----------------------- degree / norm kernels -----------------------------
__global__ void deg_init(float* deg, int n) {
    int i = blockIdx.x * blockDim.x + threadIdx.x;
    if (i < n) deg[i] = 1.0f;               // self-loop contributes 1
}
__global__ void deg_accum(const int* __restrict__ ei, float* deg, int E) {
    int e = blockIdx.x * blockDim.x + threadIdx.x;
    if (e < E) atomicAdd(&deg[ei[E + e]], 1.0f);   // dst row
}
__global__ void deg_finalize(float* dinv, int n) {
    int i = blockIdx.x * blockDim.x + threadIdx.x;
    if (i < n) dinv[i] = rsqrtf(dinv[i]);
}

// agg[i] = m[i] * dinv[i]^2  (self-loop term, also zero-initializes agg)
__global__ void agg_selfloop_init(const float* __restrict__ m,
                                  const float* __restrict__ dinv,
                                  float* __restrict__ agg, int n) {
    int idx = blockIdx.x * blockDim.x + threadIdx.x;     // n * 64 float4s
    if (idx >= n * 64) return;
    int i = idx >> 6;
    float d = dinv[i];
    float s = d * d;
    float4 v = reinterpret_cast<const float4*>(m)[idx];
    float4 o = {v.x * s, v.y * s, v.z * s, v.w * s};
    reinterpret_cast<float4*>(agg)[idx] = o;
}

// agg[dst] += m[src] * dinv[src]*dinv[dst]   (one thread per edge x 4 cols)
__global__ void scatter_edges(const int* __restrict__ ei,
                              const float* __restrict__ m,
                              const float* __restrict__ dinv,
                              float* __restrict__ agg, int E) {
    int idx = blockIdx.x * blockDim.x + threadIdx.x;     // E * 64
    if (idx >= E * 64) return;
    int e  = idx >> 6;
    int c4 = idx & 63;
    int s = ei[e];
    int d = ei[E + e];
    float nrm = dinv[s] * dinv[d];
    float4 v = reinterpret_cast<const float4*>(m + (size_t)s * 256)[c4];
    float* ap = agg + (size_t)d * 256 + c4 * 4;
    atomicAdd(ap + 0, v.x * nrm);
    atomicAdd(ap + 1, v.y * nrm);
    atomicAdd(ap + 2, v.z * nrm);
    atomicAdd(ap + 3, v.w * nrm);
}

// a = relu(a + bias)   (float4, per 256-col rows)
__global__ void relu_bias_inplace(float* __restrict__ a,
                                  const float* __restrict__ b, int n) {
    int idx = blockIdx.x * blockDim.x + threadIdx.x;     // n * 64
    if (idx >= n * 64) return;
    int c4 = idx & 63;
    float4 v  = reinterpret_cast<float4*>(a)[idx];
    float4 bb = reinterpret_cast<const float4*>(b)[c4];
    v.x = fmaxf(v.x + bb.x, 0.0f);
    v.y = fmaxf(v.y + bb.y, 0.0f);
    v.z = fmaxf(v.z + bb.z, 0.0f);
    v.w = fmaxf(v.w + bb.w, 0.0f);
    reinterpret_cast<float4*>(a)[idx] = v;
}

// GRU gates: gi[:,0:512] already holds i+h sums for r,z ; gi[:,512:768] = i_n ;
// hn = h_n part. hout = (1-u)*tanh(i_n + r*h_n) + u*prev
__global__ void gru_elementwise(const float* __restrict__ gi,
                                const float* __restrict__ hn,
                                const float* __restrict__ prev,
                                float* __restrict__ hout, int NH) {
    int idx = blockIdx.x * blockDim.x + threadIdx.x;
    if (idx >= NH) return;
    int i = idx >> 8;
    int c = idx & 255;
    const float* g = gi + (size_t)i * 768;
    float r  = 1.0f / (1.0f + __expf(-g[c]));
    float u  = 1.0f / (1.0f + __expf(-g[c + 256]));
    float nn = tanhf(g[c + 512] + r * hn[idx]);
    hout[idx] = (1.0f - u) * nn + u * prev[idx];
}

// score[e] = sum_c h[a][c]*h[b][c]*(pw[c,0]+pw[c,1]) + pb[0]+pb[1]
__global__ __launch_bounds__(256) void score_kernel(
    const float* __restrict__ h, const int* __restrict__ eli,
    const float* __restrict__ pw, const float* __restrict__ pb,
    float* __restrict__ out, int EL) {
    int lane = threadIdx.x & 31;
    int e    = blockIdx.x * 8 + (threadIdx.x >> 5);
    if (e >= EL) return;
    int a = eli[e];
    int b = eli[EL + e];
    const float* ha = h + (size_t)a * 256;
    const float* hb = h + (size_t)b * 256;
    float acc = 0.0f;
#pragma unroll
    for (int j = 0; j < 8; ++j) {
        int c = lane + 32 * j;
        float w = pw[2 * c] + pw[2 * c + 1];
        acc += ha[c] * hb[c] * w;
    }
#pragma unroll
    for (int off = 16; off > 0; off >>= 1)
        acc += __shfl_xor(acc, off, 32);
    if (lane == 0) out[e] = acc + pb[0] + pb[1];
}

// ---------------------------------------------------------------------------
extern "C" void kernel_launch(void* const* d_in, const int* in_sizes, int n_in,
                              void* d_out, int out_size, void* d_ws, size_t ws_size,
                              hipStream_t stream) {
    const float* x      = (const float*)d_in[0];
    const int*   ei     = (const int*)  d_in[1];
    const int*   eli    = (const int*)  d_in[2];
    const float* prev0  = (const float*)d_in[3];
    const float* prev1  = (const float*)d_in[4];
    const float* pre_w  = (const float*)d_in[5];
    const float* pre_b  = (const float*)d_in[6];
    const float* conv_w = (const float*)d_in[7];
    const float* conv_b = (const float*)d_in[8];
    const float* gwih   = (const float*)d_in[9];
    const float* gwhh   = (const float*)d_in[10];
    const float* gbih   = (const float*)d_in[11];
    const float* gbhh   = (const float*)d_in[12];
    const float* pw     = (const float*)d_in[13];
    const float* pb     = (const float*)d_in[14];
    float* out = (float*)d_out;

    const int DIN = 128, H = 256;
    const int N  = in_sizes[0] / DIN;   // 50000  (divisible by 80)
    const int E  = in_sizes[1] / 2;     // 800000
    const int EL = in_sizes[2] / 2;     // 200000

    // workspace layout (floats): dinv[N] | h[N*H] | gi[N*3H] (also m) | agg[N*H] (also h_n)
    float* ws   = (float*)d_ws;
    float* dinv = ws;
    float* h    = dinv + N;
    float* gi   = h + (size_t)N * H;
    float* agg  = gi + (size_t)N * 3 * H;

    // --- degree -> dinv = rsqrt(deg) ---
    deg_init<<<(N + 255) / 256, 256, 0, stream>>>(dinv, N);
    deg_accum<<<(E + 255) / 256, 256, 0, stream>>>(ei, dinv, E);
    deg_finalize<<<(N + 255) / 256, 256, 0, stream>>>(dinv, N);

    const int MG = N / 80;                         // 625 wave M-groups, exact
    auto gblk = [&](int Ntiles) { return MG * Ntiles / 4; };   // Ntiles % 4 == 0 -> exact

    // --- pre layer: h = relu(x @ pre_w + pre_b) ---
    gemm_f32_wmma<false, true, true, false><<<gblk(H / 16), 128, 0, stream>>>(
        x, DIN, pre_w, H, pre_b, h, H, DIN, H / 16);

    for (int z = 0; z < 2; ++z) {
        const float* prev = z ? prev1 : prev0;
        const float* cw  = conv_w + (size_t)z * H * H;
        const float* cb  = conv_b + (size_t)z * H;
        const float* wih = gwih + (size_t)z * 3 * H * H;
        const float* whh = gwhh + (size_t)z * 3 * H * H;
        const float* bih = gbih + (size_t)z * 3 * H;
        const float* bhh = gbhh + (size_t)z * 3 * H;

        // m = h @ conv_w[z]   (into gi[0:N*H])
        gemm_f32_wmma<false, false, false, false><<<gblk(H / 16), 128, 0, stream>>>(
            h, H, cw, H, nullptr, gi, H, H, H / 16);
        // agg = self-loop + edge scatter
        agg_selfloop_init<<<(N * 64 + 255) / 256, 256, 0, stream>>>(gi, dinv, agg, N);
        scatter_edges<<<(E * 64 + 255) / 256, 256, 0, stream>>>(ei, gi, dinv, agg, E);
        // h_conv = relu(agg + conv_b[z])  (in place)
        relu_bias_inplace<<<(N * 64 + 255) / 256, 256, 0, stream>>>(agg, cb, N);
        // gi = h_conv @ wih^T + bih   [N, 768]
        gemm_f32_wmma<true, true, false, false><<<gblk(48), 128, 0, stream>>>(
            agg, H, wih, H, bih, gi, 3 * H, H, 48);
        // gi[:,0:512] += prev @ whh[0:512]^T + bhh[0:512]
        gemm_f32_wmma<true, true, false, true><<<gblk(32), 128, 0, stream>>>(
            prev, H, whh, H, bhh, gi, 3 * H, H, 32);
        // h_n = prev @ whh[512:768]^T + bhh[512:768]  (overwrites agg)
        gemm_f32_wmma<true, true, false, false><<<gblk(16), 128, 0, stream>>>(
            prev, H, whh + (size_t)2 * H * H, H, bhh + 2 * H, agg, H, H, 16);
        // GRU combine -> h
        gru_elementwise<<<(N * H + 255) / 256, 256, 0, stream>>>(gi, agg, prev, h, N * H);
    }

    // --- link scoring ---
    score_kernel<<<(EL + 7) / 8, 256, 0, stream>>>(h, eli, pw, pb, out, EL);
}